// ChamferDistanceL2_68487548502778
// MI455X (gfx1250) — compile-verified
//
#include <hip/hip_runtime.h>

// CDNA5 / gfx1250 Chamfer-L2 via V_WMMA_F32_16X16X4_F32.
//
// d(m,n) = |x_m|^2 + |y_n|^2 - 2 x_m . y_n
// A(16x4) = [-2x, -2y, -2z, 1]  (per query row)
// B(4x16) = [yx; yy; yz; |y|^2] (per reference column)
// => WMMA D = |y_n|^2 - 2 x.y ; running min over D; add |x|^2 at the end.
//
// Roofline: 537M point pairs, 1.5 MB input (L2/LDS resident) -> pure compute.
// WMMA does all multiplies; VALU is the limiter, so the min stream is cut to
// its floor: process y-tiles in PAIRS and fold both fresh D tiles plus the
// accumulator with a single 3-input v_min3_num_f32 -> 4 VALU ops per WMMA.

typedef __attribute__((ext_vector_type(2))) float v2f;
typedef __attribute__((ext_vector_type(8))) float v8f;

#define BATCHES 8
#define NPTS 8192
#define MPTS 8192
#define XBLOCKS_PER_BATCH 16          // 8192 / 512
#define XPTS_PER_BLOCK 512            // 8 waves * 64
#define R_TILES 4                     // 16x16 query tiles per wave -> 64 rows
#define CHUNK_PTS 4096                // y points staged per LDS chunk
#define CHUNK_TILES (CHUNK_PTS / 16)  // 256
#define LDS_FLOATS (CHUNK_TILES * 64) // 16384 floats = 64 KB
#define NWAVES_TOTAL (BATCHES * XBLOCKS_PER_BATCH * 8)  // 1024 partials

// IEEE minimumNumber without the canonicalize ops fminf() drags in.
__device__ __forceinline__ float fmin_raw(float a, float b) {
    float r;
    asm("v_min_num_f32 %0, %1, %2" : "=v"(r) : "v"(a), "v"(b));
    return r;
}

// 3-input min (VOP3): one instruction folds accumulator + two fresh D tiles.
__device__ __forceinline__ float fmin3_raw(float a, float b, float c) {
    float r;
    asm("v_min3_num_f32 %0, %1, %2, %3" : "=v"(r) : "v"(a), "v"(b), "v"(c));
    return r;
}

__global__ void __launch_bounds__(256)
chamfer_wmma_main(const float* __restrict__ xyz1,
                  const float* __restrict__ xyz2,
                  float* __restrict__ partials)
{
    __shared__ float ldsB[LDS_FLOATS];

    const int tid   = threadIdx.x;
    const int lane  = tid & 31;
    const int wave  = tid >> 5;
    const int batch = blockIdx.x / XBLOCKS_PER_BATCH;
    const int xblk  = blockIdx.x % XBLOCKS_PER_BATCH;

    const float* x1b = xyz1 + (size_t)batch * NPTS * 3;
    const float* x2b = xyz2 + (size_t)batch * MPTS * 3;

    // ---- per-wave A operands: R_TILES tiles of 16x4, A = [-2x,-2y,-2z,1] ----
    // A layout (32-bit 16x4): lanes 0-15 row M=lane, vgpr0=K0, vgpr1=K1;
    //                         lanes 16-31 row M=lane-16, vgpr0=K2, vgpr1=K3.
    const int xbase = xblk * XPTS_PER_BLOCK + wave * (R_TILES * 16);
    const int mrow  = lane & 15;
    v2f   A[R_TILES];
    float sx2 = 0.0f;                  // sum of |x|^2 over this lane's rows
#pragma unroll
    for (int r = 0; r < R_TILES; ++r) {
        const float* p = x1b + (size_t)(xbase + r * 16 + mrow) * 3;
        float px = p[0], py = p[1], pz = p[2];
        sx2 += px * px + py * py + pz * pz;
        if (lane < 16) { A[r].x = -2.0f * px; A[r].y = -2.0f * py; }
        else           { A[r].x = -2.0f * pz; A[r].y = 1.0f;       }
    }

    v8f minacc[R_TILES];
#pragma unroll
    for (int r = 0; r < R_TILES; ++r)
#pragma unroll
        for (int i = 0; i < 8; ++i) minacc[r][i] = 3.4e38f;
    v8f czero = {};                    // C = 0, kept live across the loop

    for (int chunk = 0; chunk < MPTS / CHUNK_PTS; ++chunk) {
        // ---- cooperative stage: 4096 y points -> B-operand layout in LDS ----
        // B layout (4x16): lanes 0-15 hold (K0,K1)=(yx,yy) of col n=lane%16,
        //                  lanes 16-31 hold (K2,K3)=(yz,|y|^2).
        const int cbase = chunk * CHUNK_PTS;
#pragma unroll
        for (int i = 0; i < CHUNK_PTS / 256; ++i) {
            int s = i * 256 + tid;           // slot within chunk
            int t = s >> 4, c = s & 15;
            const float* p = x2b + (size_t)(cbase + s) * 3;
            float px = p[0], py = p[1], pz = p[2];
            float y2 = px * px + py * py + pz * pz;
            *reinterpret_cast<float2*>(&ldsB[t * 64 + c * 2])        = make_float2(px, py);
            *reinterpret_cast<float2*>(&ldsB[t * 64 + (c + 16) * 2]) = make_float2(pz, y2);
        }
        __syncthreads();

        const v2f* Bp = reinterpret_cast<const v2f*>(ldsB) + lane;  // +32 per tile
        // Tile PAIRS: 1 ds_load_2addr_b64 + 2*R wmma + R*8 v_min3 per pair.
        for (int t2 = 0; t2 < CHUNK_TILES / 2; ++t2) {
            v2f b0 = Bp[t2 * 64];
            v2f b1 = Bp[t2 * 64 + 32];
#pragma unroll
            for (int r = 0; r < R_TILES; ++r) {
                v8f d0 = __builtin_amdgcn_wmma_f32_16x16x4_f32(
                    false, A[r], false, b0, (short)0, czero, false, false);
                v8f d1 = __builtin_amdgcn_wmma_f32_16x16x4_f32(
                    false, A[r], false, b1, (short)0, czero, false, false);
#pragma unroll
                for (int i = 0; i < 8; ++i)
                    minacc[r][i] = fmin3_raw(minacc[r][i], d0[i], d1[i]);
            }
        }
        __syncthreads();
    }

    // ---- reduction ----
    // D layout: vgpr v, lanes 0-15 -> row m=v; lanes 16-31 -> row m=v+8.
    // Row-min = min across the 16 lanes of each half-wave group.
    float psum = 0.0f;
#pragma unroll
    for (int r = 0; r < R_TILES; ++r) {
#pragma unroll
        for (int i = 0; i < 8; ++i) {
            float v = minacc[r][i];
            v = fmin_raw(v, __shfl_xor(v, 1, 32));
            v = fmin_raw(v, __shfl_xor(v, 2, 32));
            v = fmin_raw(v, __shfl_xor(v, 4, 32));
            v = fmin_raw(v, __shfl_xor(v, 8, 32));
            psum += v;                       // replicated across the 16-lane group
        }
    }
    psum *= (1.0f / 16.0f);                  // row-min counted 16x (once per lane in group)
    psum += 0.5f * sx2;                      // each row's |x|^2 held by 2 lanes (l, l+16)
    psum += __shfl_xor(psum, 1, 32);
    psum += __shfl_xor(psum, 2, 32);
    psum += __shfl_xor(psum, 4, 32);
    psum += __shfl_xor(psum, 8, 32);
    psum += __shfl_xor(psum, 16, 32);

    if (lane == 0)
        partials[blockIdx.x * 8 + wave] = psum;
}

__global__ void __launch_bounds__(256)
chamfer_reduce(const float* __restrict__ partials, float* __restrict__ out)
{
    __shared__ float red[256];
    float s = 0.0f;
    for (int i = threadIdx.x; i < NWAVES_TOTAL; i += 256) s += partials[i];
    red[threadIdx.x] = s;
    __syncthreads();
    for (int off = 128; off > 0; off >>= 1) {
        if (threadIdx.x < off) red[threadIdx.x] += red[threadIdx.x + off];
        __syncthreads();
    }
    if (threadIdx.x == 0)
        out[0] = red[0] * (1.0f / ((float)BATCHES * (float)NPTS));
}

extern "C" void kernel_launch(void* const* d_in, const int* in_sizes, int n_in,
                              void* d_out, int out_size, void* d_ws, size_t ws_size,
                              hipStream_t stream)
{
    const float* xyz1 = (const float*)d_in[0];   // [8, 8192, 3] f32
    const float* xyz2 = (const float*)d_in[1];   // [8, 8192, 3] f32
    float* out = (float*)d_out;                  // scalar f32
    float* ws  = (float*)d_ws;                   // >= 1024 floats of scratch

    // 128 blocks = 8 batches x 16 x-slabs; 256 thr = 8 waves; every ws slot
    // is overwritten each call, so no init pass is needed (deterministic).
    chamfer_wmma_main<<<BATCHES * XBLOCKS_PER_BATCH, 256, 0, stream>>>(xyz1, xyz2, ws);
    chamfer_reduce<<<1, 256, 0, stream>>>(ws, out);
}